// SGC2_84954453114998
// MI455X (gfx1250) — compile-verified
//
#include <hip/hip_runtime.h>

#define IN_CH   128
#define NUM_HID 16
#define OUT_CH  40

typedef __attribute__((ext_vector_type(2))) float v2f;
typedef __attribute__((ext_vector_type(8))) float v8f;

// ---------------------------------------------------------------------------
// Degree (with self loop) and D^{-1/2}
// ---------------------------------------------------------------------------
__global__ void k_deg_init(int* __restrict__ deg, int n) {
    int i = blockIdx.x * blockDim.x + threadIdx.x;
    if (i < n) deg[i] = 1;                       // self loop
}

__global__ void k_deg_count(const int* __restrict__ col, int* __restrict__ deg, int E) {
    int e = blockIdx.x * blockDim.x + threadIdx.x;
    if (e < E) atomicAdd(&deg[col[e]], 1);
}

__global__ void k_dinv(const int* __restrict__ deg, float* __restrict__ dinv, int n) {
    int i = blockIdx.x * blockDim.x + threadIdx.x;
    if (i < n) dinv[i] = rsqrtf((float)deg[i]);  // deg >= 1 always (self loop)
}

// ---------------------------------------------------------------------------
// Y = X @ W_conv^T via V_WMMA_F32_16X16X4_F32, fused with t0 = dinv .* Y
// and accumulator init acc1 = t0 (self-loop term of hop 1).
// One wave per 16-node tile. N must be a multiple of 16 (100000 = 6250*16).
//
// A (16x4 f32): lane&15 = M (node row), VGPR0/1 = K, K+1; lanes>=16 hold K+2,K+3
// B (4x16 f32): lane&15 = N (hidden col), same K striping => B[k][n]=W_conv[n][k]
// C (16x16 f32): VGPR r -> M = r + 8*(lane>>4), N = lane&15
// ---------------------------------------------------------------------------
__global__ void k_gemm1_wmma(const float* __restrict__ x,
                             const float* __restrict__ Wc,
                             const float* __restrict__ dinv,
                             float* __restrict__ t0,
                             float* __restrict__ acc1,
                             int nTiles) {
    int gtid = blockIdx.x * blockDim.x + threadIdx.x;
    int tile = gtid >> 5;                 // wave id == tile id (wave32)
    if (tile >= nTiles) return;           // wave-uniform exit: EXEC stays all-1
    int lane = threadIdx.x & 31;
    int mn   = lane & 15;                 // A: row within tile, B: hidden column
    int hi   = lane >> 4;                 // 0 or 1
    int koff = hi * 2;                    // lanes 16..31 hold K+2,K+3

    int nodeBase = tile * 16;
    const float* arow = x  + (size_t)(nodeBase + mn) * IN_CH + koff;
    const float* brow = Wc + (size_t)mn * IN_CH + koff;

    v8f c = {};
#pragma unroll
    for (int k0 = 0; k0 < IN_CH; k0 += 4) {
        v2f a = *(const v2f*)(arow + k0);
        v2f b = *(const v2f*)(brow + k0);
        c = __builtin_amdgcn_wmma_f32_16x16x4_f32(
                /*neg_a=*/false, a, /*neg_b=*/false, b,
                /*c_mod=*/(short)0, c, /*reuse_a=*/false, /*reuse_b=*/false);
    }

#pragma unroll
    for (int r = 0; r < 8; ++r) {
        int node  = nodeBase + r + hi * 8;
        float val = c[r] * dinv[node];
        size_t o  = (size_t)node * NUM_HID + mn;
        t0[o]  = val;   // scaled features for the edge gather
        acc1[o] = val;  // self-loop contribution
    }
}

// ---------------------------------------------------------------------------
// Edge scatter: acc[col] += t[row], 4 lanes per edge, float4 per lane.
// Both t and acc are 6.4 MB => fully L2-resident on MI455X (192 MB L2).
// ---------------------------------------------------------------------------
__global__ void k_prop_edges(const int* __restrict__ rows,
                             const int* __restrict__ cols,
                             const float* __restrict__ t,
                             float* __restrict__ acc, int E) {
    int tid = blockIdx.x * blockDim.x + threadIdx.x;
    int e   = tid >> 2;
    if (e >= E) return;
    int cg = (tid & 3) << 2;
    int r  = rows[e];
    int c  = cols[e];
    const float4 v = *(const float4*)(t + (size_t)r * NUM_HID + cg);
    float* dst = acc + (size_t)c * NUM_HID + cg;
    atomicAdd(dst + 0, v.x);
    atomicAdd(dst + 1, v.y);
    atomicAdd(dst + 2, v.z);
    atomicAdd(dst + 3, v.w);
}

// ---------------------------------------------------------------------------
// Between hops: t1 = dinv^2 .* acc1 (== dinv .* h1 scaled again for hop 2),
// and re-init acc2 = t1 (self-loop term).
// ---------------------------------------------------------------------------
__global__ void k_rescale(const float* __restrict__ dinv,
                          float* __restrict__ t,
                          float* __restrict__ acc, int n16) {
    int i = blockIdx.x * blockDim.x + threadIdx.x;
    if (i >= n16) return;
    float d = dinv[i >> 4];
    float v = d * d * acc[i];
    t[i]   = v;
    acc[i] = v;
}

// ---------------------------------------------------------------------------
// Epilogue: z = dinv .* acc2 + b_conv ; relu ; out = z @ W_lin^T + b_lin
// One thread per (node, out-channel). W_lin/b_lin/b_conv staged in LDS.
// ---------------------------------------------------------------------------
__global__ void k_final(const float* __restrict__ acc,
                        const float* __restrict__ dinv,
                        const float* __restrict__ b_conv,
                        const float* __restrict__ W_lin,
                        const float* __restrict__ b_lin,
                        float* __restrict__ out, int total) {
    __shared__ float sW[OUT_CH * NUM_HID];
    __shared__ float sb[OUT_CH];
    __shared__ float sbc[NUM_HID];
    for (int i = threadIdx.x; i < OUT_CH * NUM_HID; i += blockDim.x) sW[i] = W_lin[i];
    if (threadIdx.x < OUT_CH)  sb[threadIdx.x]  = b_lin[threadIdx.x];
    if (threadIdx.x < NUM_HID) sbc[threadIdx.x] = b_conv[threadIdx.x];
    __syncthreads();

    int tid = blockIdx.x * blockDim.x + threadIdx.x;
    if (tid >= total) return;
    int n = tid / OUT_CH;
    int o = tid - n * OUT_CH;

    float d = dinv[n];
    const float* arow = acc + (size_t)n * NUM_HID;
    const float* wrow = sW + o * NUM_HID;
    float s = sb[o];
#pragma unroll
    for (int k = 0; k < NUM_HID; ++k) {
        float z = fmaf(d, arow[k], sbc[k]);
        z = fmaxf(z, 0.0f);
        s = fmaf(z, wrow[k], s);
    }
    out[tid] = s;
}

// ---------------------------------------------------------------------------
extern "C" void kernel_launch(void* const* d_in, const int* in_sizes, int n_in,
                              void* d_out, int out_size, void* d_ws, size_t ws_size,
                              hipStream_t stream) {
    const float* x      = (const float*)d_in[0];
    const int*   ei     = (const int*)  d_in[1];
    const float* W_conv = (const float*)d_in[2];
    const float* b_conv = (const float*)d_in[3];
    const float* W_lin  = (const float*)d_in[4];
    const float* b_lin  = (const float*)d_in[5];
    float* out = (float*)d_out;

    const int N = in_sizes[0] / IN_CH;      // 100000
    const int E = in_sizes[1] / 2;          // 1600000
    const int* rows = ei;                   // edge_index[0] = source
    const int* cols = ei + E;               // edge_index[1] = target

    // Workspace layout
    char* ws = (char*)d_ws;
    int*   deg  = (int*)  ws;                                   // N ints
    float* dinv = (float*)(ws + (size_t)N * 4);                 // N floats
    float* bufT = (float*)(ws + (size_t)N * 8);                 // N*16 floats (scaled feats)
    float* bufA = (float*)(ws + (size_t)N * 8 + (size_t)N * NUM_HID * 4); // N*16 accum

    const int B = 256;
    // 1) degree with self loops
    k_deg_init <<<(N + B - 1) / B, B, 0, stream>>>(deg, N);
    k_deg_count<<<(E + B - 1) / B, B, 0, stream>>>(cols, deg, E);
    k_dinv     <<<(N + B - 1) / B, B, 0, stream>>>(deg, dinv, N);

    // 2) project 128 -> 16 with f32 WMMA, fused dinv scale + acc init
    int nTiles = N / 16;                    // N is a multiple of 16
    int gemmThreads = nTiles * 32;
    k_gemm1_wmma<<<(gemmThreads + B - 1) / B, B, 0, stream>>>(x, W_conv, dinv, bufT, bufA, nTiles);

    // 3) hop 1: acc += gather(t)
    k_prop_edges<<<((E * 4) + B - 1) / B, B, 0, stream>>>(rows, cols, bufT, bufA, E);

    // 4) rescale for hop 2
    int n16 = N * NUM_HID;
    k_rescale<<<(n16 + B - 1) / B, B, 0, stream>>>(dinv, bufT, bufA, n16);

    // 5) hop 2
    k_prop_edges<<<((E * 4) + B - 1) / B, B, 0, stream>>>(rows, cols, bufT, bufA, E);

    // 6) fused final scale + bias + ReLU + [N,16]x[16,40] + bias
    int total = N * OUT_CH;
    k_final<<<(total + B - 1) / B, B, 0, stream>>>(bufA, dinv, b_conv, W_lin, b_lin, out, total);
}